// PQCClassifier_57028575757006
// MI455X (gfx1250) — compile-verified
//
#include <hip/hip_runtime.h>

typedef __attribute__((ext_vector_type(2))) float v2f;
typedef __attribute__((ext_vector_type(4))) float v4f;
typedef __attribute__((ext_vector_type(8))) float v8f;

// ---------------------------------------------------------------------------
// Kernel A: one wave32 computes the batch-invariant result.
// State sim (16 complex amps in registers) -> probs -> expvals via 4 chained
// V_WMMA_F32_16X16X4_F32 -> 2-way log_softmax -> {o0,o1} into workspace.
// ---------------------------------------------------------------------------
__global__ __launch_bounds__(32) void pqc_compute_kernel(
    const float* __restrict__ th_rx,
    const float* __restrict__ th_ry,
    const float* __restrict__ th_rz,
    float* __restrict__ ws)
{
    const float tx = th_rx[0];
    const float ty = th_ry[0];
    const float tz = th_rz[0];

    // ---- 16-amplitude statevector simulation, fully in registers ----
    float re[16], im[16];
#pragma unroll
    for (int i = 0; i < 16; ++i) { re[i] = 0.0f; im[i] = 0.0f; }
    re[0] = 1.0f;

    // RX(tx) on wire 0 (MSB): pairs (r, r+8); [[c,-is],[-is,c]]
    {
        float c = __cosf(tx * 0.5f), s = __sinf(tx * 0.5f);
#pragma unroll
        for (int r = 0; r < 8; ++r) {
            float ar = re[r], ai = im[r], br = re[r + 8], bi = im[r + 8];
            re[r]     = c * ar + s * bi;  im[r]     = c * ai - s * br;
            re[r + 8] = c * br + s * ai;  im[r + 8] = c * bi - s * ar;
        }
    }
    // RY(ty) on wire 1: pairs (l*8+r, l*8+4+r); [[c,-s],[s,c]]
    {
        float c = __cosf(ty * 0.5f), s = __sinf(ty * 0.5f);
#pragma unroll
        for (int l = 0; l < 2; ++l)
#pragma unroll
            for (int r = 0; r < 4; ++r) {
                int i0 = l * 8 + r, i1 = i0 + 4;
                float ar = re[i0], ai = im[i0], br = re[i1], bi = im[i1];
                re[i0] = c * ar - s * br;  im[i0] = c * ai - s * bi;
                re[i1] = s * ar + c * br;  im[i1] = s * ai + c * bi;
            }
    }
    // RZ(tz) on wire 2: phase e^{-i tz/2} if bit==0 else e^{+i tz/2}; bit=(j>>1)&1
    {
        float ch = __cosf(tz * 0.5f), sh = __sinf(tz * 0.5f);
#pragma unroll
        for (int j = 0; j < 16; ++j) {
            float ss = ((j >> 1) & 1) ? sh : -sh;
            float ar = re[j], ai = im[j];
            re[j] = ch * ar - ss * ai;
            im[j] = ch * ai + ss * ar;
        }
    }

    float p[16];
#pragma unroll
    for (int j = 0; j < 16; ++j) p[j] = re[j] * re[j] + im[j] * im[j];

    // ---- expvals = probs · signs^T via 4 chained V_WMMA_F32_16X16X4_F32 ----
    // A(16x4 f32): k = vgpr + 2*(lane>=16).  B(4x16 f32): same k rule, N=lane&15.
    // D(16x16 f32): lane holds column N=lane&15 -> D[*,n] = expval[n] for n<4.
    const int lane = (int)(threadIdx.x & 31u);
    const int half = lane >> 4;
    const int ncol = lane & 15;

    v8f acc = {0.f, 0.f, 0.f, 0.f, 0.f, 0.f, 0.f, 0.f};
#pragma unroll
    for (int blk = 0; blk < 4; ++blk) {
        v2f a, b;
#pragma unroll
        for (int v = 0; v < 2; ++v) {
            int jlo = blk * 4 + v;               // half==0 K index
            int jhi = jlo + 2;                   // half==1 K index
            a[v] = half ? p[jhi] : p[jlo];       // constant indices -> registers
            int j = half ? jhi : jlo;
            float s = 0.0f;
            int sh = 3 - ncol;                   // wire ncol reads bit (3-ncol)
            if (sh >= 0) s = 1.0f - 2.0f * (float)((j >> sh) & 1);
            b[v] = s;                            // columns n>=4 are zero
        }
        acc = __builtin_amdgcn_wmma_f32_16x16x4_f32(
            false, a, false, b, (short)0, acc, false, false);
    }

    // lanes 0..3 hold E0..E3; gather within the wave32
    float e  = acc[0];
    float E0 = __shfl(e, 0, 32);
    float E1 = __shfl(e, 1, 32);
    float E2 = __shfl(e, 2, 32);
    float E3 = __shfl(e, 3, 32);

    float l0 = E0 + E1;
    float l1 = E2 + E3;
    float mx = fmaxf(l0, l1);
    float lse = mx + __logf(__expf(l0 - mx) + __expf(l1 - mx));

    if (lane == 0) {
        ws[0] = l0 - lse;
        ws[1] = l1 - lse;
    }
}

// ---------------------------------------------------------------------------
// Kernel B: pure broadcast of the 8-byte result over the 8 MiB output.
// Uniform scalar load of {o0,o1}; 8 x 16B NT stores per thread, grid-strided
// so each wave's stores cover contiguous 512B spans.
// ---------------------------------------------------------------------------
__global__ __launch_bounds__(256) void pqc_bcast_kernel(
    const float* __restrict__ ws, v4f* __restrict__ out, int nvec4)
{
    const float o0 = ws[0];
    const float o1 = ws[1];
    v4f val = {o0, o1, o0, o1};

    const int stride = (int)(gridDim.x * blockDim.x);
    int idx = (int)(blockIdx.x * blockDim.x + threadIdx.x);
#pragma unroll
    for (int i = 0; i < 8; ++i) {
        if (idx < nvec4) __builtin_nontemporal_store(val, out + idx);
        idx += stride;
    }
}

extern "C" void kernel_launch(void* const* d_in, const int* in_sizes, int n_in,
                              void* d_out, int out_size, void* d_ws, size_t ws_size,
                              hipStream_t stream) {
    // setup_inputs order: x (unused), theta_rx, theta_ry, theta_rz
    const float* th_rx = (const float*)d_in[1];
    const float* th_ry = (const float*)d_in[2];
    const float* th_rz = (const float*)d_in[3];
    float* ws  = (float*)d_ws;
    float* out = (float*)d_out;

    // 1) one wave computes the invariant pair (WMMA path)
    pqc_compute_kernel<<<1, 32, 0, stream>>>(th_rx, th_ry, th_rz, ws);

    // 2) stream 8 MiB of broadcast stores at HBM rate
    int nvec4 = out_size / 4;                 // out_size = 2*BSZ, divisible by 4
    const int threads = 256;                  // 8 wave32 waves per workgroup
    const int per_thread = 8;                 // 128 B per thread
    int blocks = (nvec4 + threads * per_thread - 1) / (threads * per_thread);
    pqc_bcast_kernel<<<blocks, threads, 0, stream>>>(ws, (v4f*)out, nvec4);
}